// GCN_6871947674193
// MI455X (gfx1250) — compile-verified
//
#include <hip/hip_runtime.h>
#include <hip/hip_bf16.h>
#include <cstdint>

#define N_NODES 100000
#define N_EDGES 1600000
#define IN_DIM  128
#define OUT_DIM 64
#define ROW_TILES (N_NODES / 16)   // 6250, exact

typedef __attribute__((ext_vector_type(2))) float v2f;
typedef __attribute__((ext_vector_type(8))) float v8f;

// ---------------------------------------------------------------------------
// Kernel 0: out <- bias broadcast, deg <- 0   (harness poisons d_out/d_ws)
// ---------------------------------------------------------------------------
__global__ __launch_bounds__(256) void gcn_init(float* __restrict__ out,
                                                const float* __restrict__ bias,
                                                float* __restrict__ deg) {
    int idx = blockIdx.x * blockDim.x + threadIdx.x;
    if (idx < N_NODES * OUT_DIM) out[idx] = bias[idx & (OUT_DIM - 1)];
    if (idx < N_NODES)           deg[idx] = 0.0f;
}

// ---------------------------------------------------------------------------
// Kernel 1: deg[col[e]] += 1   (unit edge weights, PyG gcn_norm)
// ---------------------------------------------------------------------------
__global__ __launch_bounds__(256) void gcn_degree(const long long* __restrict__ col,
                                                  float* __restrict__ deg) {
    int e = blockIdx.x * blockDim.x + threadIdx.x;
    if (e < N_EDGES) atomicAdd(&deg[(int)col[e]], 1.0f);
}

// ---------------------------------------------------------------------------
// Kernel 2: dinv[i] = deg>0 ? rsqrt(deg) : 0   (deg integral => max(deg,1)=deg)
// ---------------------------------------------------------------------------
__global__ __launch_bounds__(256) void gcn_dinv(const float* __restrict__ deg,
                                                float* __restrict__ dinv) {
    int i = blockIdx.x * blockDim.x + threadIdx.x;
    if (i < N_NODES) {
        float d = deg[i];
        dinv[i] = (d > 0.0f) ? rsqrtf(d) : 0.0f;
    }
}

// ---------------------------------------------------------------------------
// Kernel 3: Xp = X @ W via V_WMMA_F32_16X16X4_F32 (full f32 accuracy).
//
// 256 threads = 8 wave32, one 16-row tile of X per wave, 4 C tiles (v8f)
// covering all 64 output cols. W staged in LDS pre-swizzled so each B
// fragment (4x16, f32) is one aligned ds_load_b64 per lane:
//   lanes 0-15  hold K = {2p,   2p+1}, N = lane
//   lanes 16-31 hold K = {2p+2, 2p+3}, N = lane-16
// A fragment (16x4, f32) is one global_load_b64 per lane:
//   lanes 0-15:  row = base+lane,    K = {k0,   k0+1}
//   lanes 16-31: row = base+lane-16, K = {k0+2, k0+3}
// ---------------------------------------------------------------------------
__global__ __launch_bounds__(256) void gcn_gemm(const float* __restrict__ X,
                                                const float* __restrict__ W,
                                                float* __restrict__ Xp) {
    __shared__ float wlds[IN_DIM * OUT_DIM];   // 32 KB, K-pair swizzled

    // Fill LDS: wlds[((k>>1)*64 + n)*2 + (k&1)] = W[k*64 + n]
    for (int i = threadIdx.x; i < IN_DIM * OUT_DIM; i += 256) {
        int k = i >> 6;        // row of W (K)
        int n = i & 63;        // col of W (N)
        wlds[(((k >> 1) << 6) + n) * 2 + (k & 1)] = W[i];
    }
    __syncthreads();

    const int wave   = threadIdx.x >> 5;
    const int lane   = threadIdx.x & 31;
    const int tile   = blockIdx.x * 8 + wave;
    if (tile >= ROW_TILES) return;             // wave-uniform: EXEC stays all-1s

    const int laneHi = lane >> 4;              // 0 or 1 (K half)
    const int laneLo = lane & 15;              // M (A) / N-in-tile (B)
    const int rowBase = tile * 16;

    const float* aPtr = X + (size_t)(rowBase + laneLo) * IN_DIM + 2 * laneHi;

    v8f c0 = {}, c1 = {}, c2 = {}, c3 = {};

    #pragma unroll 4
    for (int k0 = 0; k0 < IN_DIM; k0 += 4) {
        v2f a = *(const v2f*)(aPtr + k0);                  // global_load_b64

        const int p = (k0 >> 1) + laneHi;                  // K-pair index
        const float* bbase = wlds + (p << 7);              // p*64*2 floats
        v2f b0 = *(const v2f*)(bbase + ((0 * 16 + laneLo) << 1));  // ds_load_b64
        v2f b1 = *(const v2f*)(bbase + ((1 * 16 + laneLo) << 1));
        v2f b2 = *(const v2f*)(bbase + ((2 * 16 + laneLo) << 1));
        v2f b3 = *(const v2f*)(bbase + ((3 * 16 + laneLo) << 1));

        c0 = __builtin_amdgcn_wmma_f32_16x16x4_f32(false, a, false, b0, (short)0, c0, false, false);
        c1 = __builtin_amdgcn_wmma_f32_16x16x4_f32(false, a, false, b1, (short)0, c1, false, false);
        c2 = __builtin_amdgcn_wmma_f32_16x16x4_f32(false, a, false, b2, (short)0, c2, false, false);
        c3 = __builtin_amdgcn_wmma_f32_16x16x4_f32(false, a, false, b3, (short)0, c3, false, false);
    }

    // C/D layout: VGPR v, lanes 0-15 -> M=v, N=lane; lanes 16-31 -> M=v+8, N=lane-16
    float* obase = Xp + (size_t)(rowBase + 8 * laneHi) * OUT_DIM + laneLo;
    #pragma unroll
    for (int v = 0; v < 8; ++v) {
        obase[(size_t)v * OUT_DIM +  0] = c0[v];
        obase[(size_t)v * OUT_DIM + 16] = c1[v];
        obase[(size_t)v * OUT_DIM + 32] = c2[v];
        obase[(size_t)v * OUT_DIM + 48] = c3[v];
    }
}

// ---------------------------------------------------------------------------
// Kernel 4: edge scatter. One wave32 per edge; each lane handles 2 features
// (float2 load of Xp row + 2x global_atomic_add_f32 into out[col]).
// out (25.6 MB) fits in the 192 MB L2, so atomics resolve at L2.
// ---------------------------------------------------------------------------
__global__ __launch_bounds__(256) void gcn_scatter(const long long* __restrict__ row,
                                                   const long long* __restrict__ col,
                                                   const float* __restrict__ Xp,
                                                   const float* __restrict__ dinv,
                                                   float* __restrict__ out) {
    const int e    = (blockIdx.x * blockDim.x + threadIdx.x) >> 5;  // wave-uniform
    const int lane = threadIdx.x & 31;
    if (e >= N_EDGES) return;

    const int r = (int)row[e];
    const int c = (int)col[e];
    const float nrm = dinv[r] * dinv[c];

    const float* src = Xp  + (size_t)r * OUT_DIM + 2 * lane;
    float*       dst = out + (size_t)c * OUT_DIM + 2 * lane;

    v2f v = *(const v2f*)src;                 // global_load_b64
    atomicAdd(dst + 0, v.x * nrm);            // global_atomic_add_f32
    atomicAdd(dst + 1, v.y * nrm);
}

// ---------------------------------------------------------------------------
// Launch
// ---------------------------------------------------------------------------
extern "C" void kernel_launch(void* const* d_in, const int* in_sizes, int n_in,
                              void* d_out, int out_size, void* d_ws, size_t ws_size,
                              hipStream_t stream) {
    const float*     X     = (const float*)d_in[0];         // [N, 128] f32
    const long long* edges = (const long long*)d_in[1];     // [2, E] int64
    const float*     W     = (const float*)d_in[2];         // [128, 64] f32
    const float*     bias  = (const float*)d_in[3];         // [64] f32
    float*           out   = (float*)d_out;                 // [N, 64] f32

    float* Xp   = (float*)d_ws;                             // N*64 floats
    float* deg  = Xp  + (size_t)N_NODES * OUT_DIM;          // N floats
    float* dinv = deg + N_NODES;                            // N floats

    const long long* rowIdx = edges;                        // sources
    const long long* colIdx = edges + N_EDGES;              // targets

    gcn_init   <<<(N_NODES * OUT_DIM + 255) / 256, 256, 0, stream>>>(out, bias, deg);
    gcn_degree <<<(N_EDGES + 255) / 256,           256, 0, stream>>>(colIdx, deg);
    gcn_dinv   <<<(N_NODES + 255) / 256,           256, 0, stream>>>(deg, dinv);
    gcn_gemm   <<<(ROW_TILES + 7) / 8,             256, 0, stream>>>(X, W, Xp);

    const long long scatterThreads = (long long)N_EDGES * 32;
    gcn_scatter<<<(unsigned)((scatterThreads + 255) / 256), 256, 0, stream>>>(
        rowIdx, colIdx, Xp, dinv, out);
}